// CriterionIFV_14766097563884
// MI455X (gfx1250) — compile-verified
//
#include <hip/hip_runtime.h>
#include <hip/hip_bf16.h>
#include <math.h>

typedef __attribute__((ext_vector_type(2))) float v2f;
typedef __attribute__((ext_vector_type(8))) float v8f;

#define KCLS 19
#define CCH  512
#define NPIX 4096   // 64*64
#define NB   8
#define NSEG 4      // pixel segments for k2, channel segments for k4

// ---------------- workspace layout (bytes) ----------------
// lab   : int [8*4096]                          @ 0        (131072)
// cnt   : int [8*19]                            @ 131072   (608, pad)
// sp    : f32 [4 seg][8][2][19][512] partials   @ 132096   (2490368)
// means : f32 [8][2][19][512]                   @ 2622464  (622592)
// nrm   : f32 [8][2][19]                        @ 3245056  (1216, pad)
// part  : f32 [512]                             @ 3246336  (2048)
#define OFF_LAB   0u
#define OFF_CNT   131072u
#define OFF_SP    132096u
#define OFF_MEANS 2622464u
#define OFF_NORM  3245056u
#define OFF_PART  3246336u

#define SPROW ((size_t)NB * 2 * KCLS * CCH)   // one segment's sums, in floats

// ---------------- k0: zero counts ----------------
__global__ __launch_bounds__(256) void k0_zero(int* __restrict__ cnt) {
    int t = threadIdx.x;
    if (t < NB * KCLS) cnt[t] = 0;
}

// ---------------- k1: nearest-resize labels + counts ----------------
__global__ __launch_bounds__(256) void k1_labels(const int* __restrict__ tgt,
                                                 int* __restrict__ lab,
                                                 int* __restrict__ cnt) {
    int bx  = blockIdx.x;                       // 128 blocks
    int b   = bx >> 4;
    int idx = ((bx & 15) << 8) + threadIdx.x;   // 0..4095
    int h = idx >> 6, w = idx & 63;
    // nearest: in_idx = floor(out*512/64) = 8*out
    int v = tgt[((size_t)b * 512 + (size_t)h * 8) * 512 + (size_t)w * 8];
    lab[b * NPIX + idx] = v;
    if (v >= 0 && v < KCLS) atomicAdd(&cnt[b * KCLS + v], 1);
}

// ---------------- k2: one-hot GEMM via V_WMMA_F32_16X16X4_F32 ----------------
// grid = (32 ch-tiles, 8 batches, 2 tensors * 4 pixel-segments) = 2048 blocks
// block = 128 (4 waves). Register double-buffered LDS staging, b128 loads.
__global__ __launch_bounds__(128) void k2_class_sums(const float* __restrict__ S,
                                                     const float* __restrict__ T,
                                                     const int* __restrict__ lab,
                                                     float* __restrict__ sp) {
    const int chTile = blockIdx.x;               // 16 channels
    const int b      = blockIdx.y;
    const int tz     = blockIdx.z;
    const int t      = tz & 1;
    const int seg    = tz >> 1;                  // 1024 pixels per segment
    const float* __restrict__ F =
        (t == 0 ? S : T) + ((size_t)b * CCH + (size_t)chTile * 16) * NPIX + seg * 1024;
    const int* __restrict__ labB = lab + b * NPIX + seg * 1024;

    __shared__ float tileF[16 * 132];            // stride 132: bank-conflict-free
    __shared__ int   labT[128];
    __shared__ float accb[4][16][32];            // per-wave exclusive accumulators

    const int tid    = threadIdx.x;
    const int wave   = tid >> 5;
    const int lane   = tid & 31;
    const int lane16 = lane & 15;
    const int half   = lane >> 4;                // 0: K=0,1  1: K=2,3

    // staging addresses: pass j covers rows j*4 + (tid>>5), cols (tid&31)*4 (+3)
    const int srow = tid >> 5;
    const int scol = (tid & 31) * 4;

    v8f acc0 = {};                               // classes 0..15
    v8f acc1 = {};                               // classes 16..31 (19..31 stay 0)

    float4 rbuf[4];
    int    rlab;
    #pragma unroll
    for (int j = 0; j < 4; ++j)                  // prefetch chunk 0 into regs
        rbuf[j] = *(const float4*)&F[(size_t)(j * 4 + srow) * NPIX + scol];
    rlab = labB[tid];

    for (int chunk = 0; chunk < 1024 / 128; ++chunk) {
        #pragma unroll
        for (int j = 0; j < 4; ++j)              // regs -> LDS
            *(float4*)&tileF[(j * 4 + srow) * 132 + scol] = rbuf[j];
        labT[tid] = rlab;
        __syncthreads();

        if (chunk < 7) {                         // overlap next chunk's loads
            const int n1 = (chunk + 1) * 128;    // with this chunk's WMMAs
            #pragma unroll
            for (int j = 0; j < 4; ++j)
                rbuf[j] = *(const float4*)&F[(size_t)(j * 4 + srow) * NPIX + n1 + scol];
            rlab = labB[n1 + tid];
        }

        const int pbase = wave * 32;             // this wave's 32 pixels
        #pragma unroll
        for (int s = 0; s < 8; ++s) {            // K=4 pixels per step
            const int q = pbase + s * 4 + half * 2;
            v2f A;                               // A[M=channel lane16][K]
            A.x = tileF[lane16 * 132 + q];
            A.y = tileF[lane16 * 132 + q + 1];
            const int l0 = labT[q];
            const int l1 = labT[q + 1];
            v2f B0, B1;                          // one-hot B[K=pixel][N=class]
            B0.x = (l0 == lane16)      ? 1.0f : 0.0f;
            B0.y = (l1 == lane16)      ? 1.0f : 0.0f;
            B1.x = (l0 == lane16 + 16) ? 1.0f : 0.0f;
            B1.y = (l1 == lane16 + 16) ? 1.0f : 0.0f;
            acc0 = __builtin_amdgcn_wmma_f32_16x16x4_f32(false, A, false, B0,
                                                         (short)0, acc0, false, false);
            acc1 = __builtin_amdgcn_wmma_f32_16x16x4_f32(false, A, false, B1,
                                                         (short)0, acc1, false, false);
        }
        __syncthreads();
    }

    // D layout: VGPR r, lanes0-15 -> M=r, lanes16-31 -> M=r+8; N = lane&15
    #pragma unroll
    for (int r = 0; r < 8; ++r) {
        const int M = r + half * 8;
        accb[wave][M][lane16]      = acc0[r];
        accb[wave][M][lane16 + 16] = acc1[r];
    }
    __syncthreads();

    for (int e = tid; e < 16 * 32; e += 128) {   // deterministic 4-wave sum
        const int M = e >> 5, k = e & 31;
        if (k < KCLS) {
            float v = accb[0][M][k] + accb[1][M][k] + accb[2][M][k] + accb[3][M][k];
            sp[(size_t)seg * SPROW +
               (((size_t)b * 2 + t) * KCLS + k) * CCH + chTile * 16 + M] = v;
        }
    }
}

// ---------------- k3: combine segments -> means + mean norms ----------------
// grid = 8*2*19 rows, block = 128
__global__ __launch_bounds__(128) void k3_means(const float* __restrict__ sp,
                                                const int* __restrict__ cnt,
                                                float* __restrict__ means,
                                                float* __restrict__ nrm) {
    const int row = blockIdx.x;                  // (b*2+t)*19 + k
    const int k   = row % KCLS;
    const int b   = (row / KCLS) >> 1;
    const float inv = 1.0f / ((float)cnt[b * KCLS + k] + 1e-6f);
    const float* s0 = sp + (size_t)row * CCH;
    float* m = means + (size_t)row * CCH;
    float ss = 0.0f;
    for (int i = threadIdx.x; i < CCH; i += 128) {
        float v = (s0[i] + s0[SPROW + i] + s0[2 * SPROW + i] + s0[3 * SPROW + i]) * inv;
        m[i] = v;
        ss += v * v;
    }
    __shared__ float red[128];
    red[threadIdx.x] = ss;
    __syncthreads();
    for (int o = 64; o > 0; o >>= 1) {
        if (threadIdx.x < o) red[threadIdx.x] += red[threadIdx.x + o];
        __syncthreads();
    }
    if (threadIdx.x == 0) nrm[row] = sqrtf(red[0]);
}

// ---------------- k4: cosines + MSE partials ----------------
// grid = 512 blocks (64 pixels each), block = 256 = 64 px * 4 channel-segments
__global__ __launch_bounds__(256) void k4_cos(const float* __restrict__ S,
                                              const float* __restrict__ T,
                                              const float* __restrict__ means,
                                              const float* __restrict__ nrm,
                                              const int* __restrict__ lab,
                                              float* __restrict__ part) {
    const int bx = blockIdx.x;
    const int b  = bx >> 6;                      // 64 blocks per batch
    const int px = threadIdx.x & 63;             // pixel within tile
    const int cs = threadIdx.x >> 6;             // channel segment (uniform per wave)
    const int p  = (bx & 63) * 64 + px;
    int l = lab[b * NPIX + p];
    l = l < 0 ? 0 : (l >= KCLS ? KCLS - 1 : l);

    const float* __restrict__ fS = S + ((size_t)b * CCH + cs * 128) * NPIX + p;
    const float* __restrict__ fT = T + ((size_t)b * CCH + cs * 128) * NPIX + p;
    const float* __restrict__ mS = means + ((size_t)(b * 2 + 0) * KCLS + l) * CCH + cs * 128;
    const float* __restrict__ mT = means + ((size_t)(b * 2 + 1) * KCLS + l) * CCH + cs * 128;

    float numS = 0.f, sqS = 0.f, numT = 0.f, sqT = 0.f;
    #pragma unroll 8
    for (int c = 0; c < 128; ++c) {
        float a  = fS[(size_t)c * NPIX];
        float m  = mS[c];
        numS += a * m;  sqS += a * a;
        float at = fT[(size_t)c * NPIX];
        float mt = mT[c];
        numT += at * mt; sqT += at * at;
    }

    __shared__ float red4[NSEG][64][4];
    red4[cs][px][0] = numS; red4[cs][px][1] = sqS;
    red4[cs][px][2] = numT; red4[cs][px][3] = sqT;
    __syncthreads();

    __shared__ float dd[64];
    if (threadIdx.x < 64) {                      // fixed-order 4-segment combine
        float nS = 0.f, qS = 0.f, nT = 0.f, qT = 0.f;
        #pragma unroll
        for (int s = 0; s < NSEG; ++s) {
            nS += red4[s][threadIdx.x][0]; qS += red4[s][threadIdx.x][1];
            nT += red4[s][threadIdx.x][2]; qT += red4[s][threadIdx.x][3];
        }
        const float rS = nrm[(b * 2 + 0) * KCLS + l];
        const float rT = nrm[(b * 2 + 1) * KCLS + l];
        const float cSim = nS / (fmaxf(sqrtf(qS), 1e-8f) * fmaxf(rS, 1e-8f));
        const float cTim = nT / (fmaxf(sqrtf(qT), 1e-8f) * fmaxf(rT, 1e-8f));
        const float d = cSim - cTim;
        dd[threadIdx.x] = d * d;
    }
    __syncthreads();
    for (int o = 32; o > 0; o >>= 1) {
        if (threadIdx.x < o) dd[threadIdx.x] += dd[threadIdx.x + o];
        __syncthreads();
    }
    if (threadIdx.x == 0) part[bx] = dd[0];
}

// ---------------- k5: final fixed-order reduction over 512 partials ----------------
__global__ __launch_bounds__(256) void k5_final(const float* __restrict__ part,
                                                float* __restrict__ out) {
    __shared__ float red[256];
    red[threadIdx.x] = part[threadIdx.x] + part[threadIdx.x + 256];
    __syncthreads();
    for (int o = 128; o > 0; o >>= 1) {
        if (threadIdx.x < o) red[threadIdx.x] += red[threadIdx.x + o];
        __syncthreads();
    }
    if (threadIdx.x == 0) out[0] = red[0] / (float)(NB * NPIX);
}

extern "C" void kernel_launch(void* const* d_in, const int* in_sizes, int n_in,
                              void* d_out, int out_size, void* d_ws, size_t ws_size,
                              hipStream_t stream) {
    (void)in_sizes; (void)n_in; (void)out_size; (void)ws_size;
    const float* S   = (const float*)d_in[0];
    const float* T   = (const float*)d_in[1];
    const int*   tgt = (const int*)d_in[2];
    char* ws = (char*)d_ws;
    int*   lab   = (int*)(ws + OFF_LAB);
    int*   cnt   = (int*)(ws + OFF_CNT);
    float* sp    = (float*)(ws + OFF_SP);
    float* means = (float*)(ws + OFF_MEANS);
    float* nrm   = (float*)(ws + OFF_NORM);
    float* part  = (float*)(ws + OFF_PART);
    float* out   = (float*)d_out;

    k0_zero<<<1, 256, 0, stream>>>(cnt);
    k1_labels<<<128, 256, 0, stream>>>(tgt, lab, cnt);
    k2_class_sums<<<dim3(32, NB, 2 * NSEG), 128, 0, stream>>>(S, T, lab, sp);
    k3_means<<<NB * 2 * KCLS, 128, 0, stream>>>(sp, cnt, means, nrm);
    k4_cos<<<512, 256, 0, stream>>>(S, T, means, nrm, lab, part);
    k5_final<<<1, 256, 0, stream>>>(part, out);
}